// kNN_70119636075070
// MI455X (gfx1250) — compile-verified
//
#include <hip/hip_runtime.h>
#include <math.h>

typedef __attribute__((ext_vector_type(2))) float v2f;
typedef __attribute__((ext_vector_type(8))) float v8f;

#define BATCH 2
#define NQ    4096
#define NR    16384
#define DIM   128
#define KSEL  16

#define ROWSTRIDE 132                 // 128 + 4 floats pad -> conflict-free ds_load_b64
#define BUFFLOATS (16 * ROWSTRIDE)    // 2112 floats = 8448 B per buffer

// ---------------------------------------------------------------------------
// CDNA5 async global->LDS copy (ASYNCcnt), inline asm (portable per ISA doc).
// ---------------------------------------------------------------------------
__device__ __forceinline__ void async_load_b128(const void* gaddr, unsigned ldsOff) {
  asm volatile("global_load_async_to_lds_b128 %0, %1, off"
               :: "v"(ldsOff), "v"(gaddr) : "memory");
}
__device__ __forceinline__ void wait_asynccnt_le4() {
  asm volatile("s_wait_asynccnt 0x4" ::: "memory");
}

// ---------------------------------------------------------------------------
// Prep: row norms ||r_i||^2 and ||q_i||^2 into workspace.
// ---------------------------------------------------------------------------
__global__ void knn_prep_norms(const float* __restrict__ q,
                               const float* __restrict__ r,
                               float* __restrict__ rn,
                               float* __restrict__ qn) {
  int i = blockIdx.x * blockDim.x + threadIdx.x;
  const int nrTot = BATCH * NR;
  const int nqTot = BATCH * NQ;
  if (i < nrTot) {
    const float4* p = (const float4*)(r + (size_t)i * DIM);
    float s = 0.f;
#pragma unroll
    for (int k = 0; k < DIM / 4; ++k) {
      float4 v = p[k];
      s += v.x * v.x + v.y * v.y + v.z * v.z + v.w * v.w;
    }
    rn[i] = s;
  } else if (i < nrTot + nqTot) {
    int j = i - nrTot;
    const float4* p = (const float4*)(q + (size_t)j * DIM);
    float s = 0.f;
#pragma unroll
    for (int k = 0; k < DIM / 4; ++k) {
      float4 v = p[k];
      s += v.x * v.x + v.y * v.y + v.z * v.z + v.w * v.w;
    }
    qn[j] = s;
  }
}

// ---------------------------------------------------------------------------
// Fused pairwise-L2 + top-16.
// Block = 4 waves, all sweeping the same r-column tiles (shared async-staged
// LDS buffer), each wave owning its own 16-query tile.
// ---------------------------------------------------------------------------
__global__ void __launch_bounds__(128) knn_wmma_topk(
    const float* __restrict__ q, const float* __restrict__ r,
    const float* __restrict__ qn, const float* __restrict__ rn,
    int* __restrict__ outIdx, float* __restrict__ outD) {
  __shared__ float rbuf[2][BUFFLOATS];  // block-shared, double-buffered r tile
  __shared__ float sTile[4][16 * 16];   // per-wave transposed score tile
  __shared__ float sMD[4][16 * 32];     // per-wave merge distances
  __shared__ int   sMI[4][16 * 32];     // per-wave merge indices

  const int tid  = threadIdx.x;
  const int lane = tid & 31;
  const int wave = tid >> 5;
  const int b    = blockIdx.x >> 6;                 // batch
  const int m0   = ((blockIdx.x & 63) * 4 + wave) << 4;  // query tile base
  const int mq   = lane & 15;
  const int h    = lane >> 4;

  // ---- per-thread async-copy offsets (loop-invariant) ---------------------
  // tile = 8KB contiguous: 4 issues x 128 threads x 16B. LDS dest is padded.
  unsigned goff[4], loff[4];
#pragma unroll
  for (int i = 0; i < 4; ++i) {
    unsigned o = (unsigned)i * 2048u + (unsigned)tid * 16u;  // 0..8191
    goff[i] = o;
    loff[i] = (o >> 9) * (ROWSTRIDE * 4) + (o & 511);        // row*528 + within
  }
  const unsigned ldsBase0 = (unsigned)(uintptr_t)&rbuf[0][0];
  const unsigned ldsBase1 = (unsigned)(uintptr_t)&rbuf[1][0];
  const char* rBatch = (const char*)(r + (size_t)b * NR * DIM);

  // ---- A fragments (q tile), loop-invariant, in registers -----------------
  const float* qrow = q + ((size_t)b * NQ + m0 + mq) * DIM + h * 2;
  v2f a[32];
#pragma unroll
  for (int kk = 0; kk < 32; ++kk)
    a[kk] = *(const v2f*)(qrow + kk * 4);

  // ---- register-resident top-16 (half the columns per lane) ---------------
  float bestD[16];
  int   bestI[16];
#pragma unroll
  for (int t = 0; t < 16; ++t) { bestD[t] = INFINITY; bestI[t] = -1; }
  float worst = INFINITY;
  int   worstPos = 0;

  const float* rnb = rn + b * NR;
  const int nlo = h * 8;

  // ---- prologue: stage tile 0 into buffer 0 -------------------------------
#pragma unroll
  for (int i = 0; i < 4; ++i)
    async_load_b128(rBatch + goff[i], ldsBase0 + loff[i]);

  int cur = 0;
  for (int n0 = 0; n0 < NR; n0 += 16) {
    // issue next tile into the other buffer (clamped reload on last iter)
    const int nn = (n0 + 16 < NR) ? (n0 + 16) : n0;
    const unsigned dstBase = cur ? ldsBase0 : ldsBase1;
    const char* gsrc = rBatch + (size_t)nn * (DIM * 4);
#pragma unroll
    for (int i = 0; i < 4; ++i)
      async_load_b128(gsrc + goff[i], dstBase + loff[i]);

    wait_asynccnt_le4();   // this wave's share of the current tile is done
    __syncthreads();       // all waves' shares visible

    float rnv = rnb[n0 + mq];   // issued early; consumed after the WMMA block

    // ---- 32 x V_WMMA_F32_16X16X4_F32 from LDS, ping-pong B fragments ------
    const float* lb = &rbuf[cur][mq * ROWSTRIDE + h * 2];
    v8f c = {};
    v2f p0[8], p1[8];
#pragma unroll
    for (int j = 0; j < 8; ++j) p0[j] = *(const v2f*)(lb + j * 4);
#pragma unroll
    for (int g = 0; g < 4; ++g) {
      v2f* curf = (g & 1) ? p1 : p0;
      v2f* nxtf = (g & 1) ? p0 : p1;
      if (g < 3) {
#pragma unroll
        for (int j = 0; j < 8; ++j)
          nxtf[j] = *(const v2f*)(lb + ((g + 1) * 8 + j) * 4);
      }
#pragma unroll
      for (int j = 0; j < 8; ++j)
        c = __builtin_amdgcn_wmma_f32_16x16x4_f32(
            false, a[g * 8 + j], false, curf[j], (short)0, c, false, false);
    }

    // s = ||r_n||^2 - 2 q.r  (qn added only at the end; rank-invariant)
#pragma unroll
    for (int v = 0; v < 8; ++v)
      sTile[wave][(v + 8 * h) * 16 + mq] = fmaf(-2.f, c[v], rnv);
    __builtin_amdgcn_wave_barrier();             // same-wave LDS is in-order

    // lane (mq,h) scans 8 candidates of query mq
#pragma unroll
    for (int j = 0; j < 8; ++j) {
      float s = sTile[wave][mq * 16 + nlo + j];
      if (s < worst) {                           // rare replace path
        int col = n0 + nlo + j;
#pragma unroll
        for (int t = 0; t < 16; ++t)
          if (t == worstPos) { bestD[t] = s; bestI[t] = col; }
        worst = bestD[0]; worstPos = 0;
#pragma unroll
        for (int t = 1; t < 16; ++t)
          if (bestD[t] > worst) { worst = bestD[t]; worstPos = t; }
      }
    }

    __syncthreads();       // everyone done reading buf[cur] before overwrite
    cur ^= 1;
  }

  // ---- merge the two half-lists per query, emit sorted top-16 -------------
#pragma unroll
  for (int t = 0; t < 16; ++t) {
    sMD[wave][mq * 32 + h * 16 + t] = bestD[t];
    sMI[wave][mq * 32 + h * 16 + t] = bestI[t];
  }
  __builtin_amdgcn_wave_barrier();

  if (h == 0) {
    float qnm = qn[b * NQ + m0 + mq];
    float* md = &sMD[wave][mq * 32];
    int*   mi = &sMI[wave][mq * 32];
    size_t base = ((size_t)b * NQ + m0 + mq) * KSEL;
    for (int k = 0; k < KSEL; ++k) {
      float best = INFINITY; int bi = 0;
#pragma unroll
      for (int t = 0; t < 32; ++t) {
        float v = md[t];
        if (v < best) { best = v; bi = t; }
      }
      outIdx[base + k] = mi[bi];
      outD[base + k]   = sqrtf(fmaxf(qnm + best, 0.f));
      md[bi] = INFINITY;
    }
  }
}

// ---------------------------------------------------------------------------
extern "C" void kernel_launch(void* const* d_in, const int* in_sizes, int n_in,
                              void* d_out, int out_size, void* d_ws, size_t ws_size,
                              hipStream_t stream) {
  const float* q = (const float*)d_in[0];   // [2,4096,128]
  const float* r = (const float*)d_in[1];   // [2,16384,128]

  float* rn = (float*)d_ws;                 // BATCH*NR floats
  float* qn = rn + (size_t)BATCH * NR;      // BATCH*NQ floats

  int*   outIdx = (int*)d_out;                                   // [2,4096,16] int32
  float* outD   = (float*)d_out + (size_t)BATCH * NQ * KSEL;     // [2,4096,16] f32

  const int totalRows = BATCH * NR + BATCH * NQ;                 // 40960
  knn_prep_norms<<<(totalRows + 255) / 256, 256, 0, stream>>>(q, r, rn, qn);

  // 512 query tiles = 128 blocks x 4 waves; 4 tiles share one r-tile stream.
  knn_wmma_topk<<<128, 128, 0, stream>>>(q, r, qn, rn, outIdx, outD);
}